// BotanAHT_8959301779559
// MI455X (gfx1250) — compile-verified
//
#include <hip/hip_runtime.h>

// ---------------------------------------------------------------------------
// Block-Hadamard (HS=128) via Kronecker split H128 = H8 (x) H16:
//   - H8 along the chunk index: 3 in-register butterfly stages (VALU adds)
//   - H16 along the element index: 4 chained v_wmma_f32_16x16x4_f32 (K=16)
// Each wave handles 2 blocks (256 floats) per iteration; 16x16 transpose
// between the two phases goes through wave-private LDS (pad-17, conflict-free).
// Memory-bound: 512 MB total traffic -> ~22 us floor @ 23.3 TB/s; matrix work
// is ~1M f32 WMMAs chip-wide, a few us. Streaming -> nontemporal global ops.
// ---------------------------------------------------------------------------

typedef float v2f __attribute__((ext_vector_type(2)));
typedef float v8f __attribute__((ext_vector_type(8)));

#define HS 128
#define PAD 17  // 16 + 1 padding word: conflict-free b32 LDS access

__global__ __launch_bounds__(256) void fht128_wmma_kernel(
    const float* __restrict__ x, const float* __restrict__ H,
    float* __restrict__ y, unsigned int total_pairs) {
  // 8 waves per workgroup, each with a private 16x17-word LDS tile
  __shared__ float lds[8 * 16 * PAD];

  const unsigned lane = threadIdx.x & 31u;
  const unsigned wave = threadIdx.x >> 5;
  const unsigned e  = lane & 15u;  // element-in-chunk (N index / A row index)
  const unsigned hi = lane >> 4;   // half-wave: block select & K-pair select
  float* wl = &lds[wave * (16 * PAD)];

  // ---- B fragments: upper-left 16x16 of H IS Sylvester H16 * (1/sqrt(128)).
  // f32 WMMA B 4x16 layout: VGPR0 = row 4j+0 (lanes 0-15) / row 4j+2
  // (lanes 16-31), VGPR1 = rows 4j+1 / 4j+3; N = lane&15. Loaded once.
  v2f bfrag[4];
#pragma unroll
  for (int j = 0; j < 4; ++j) {
    const unsigned r0 = 4u * (unsigned)j + 2u * hi;
    bfrag[j].x = H[r0 * HS + e];
    bfrag[j].y = H[(r0 + 1u) * HS + e];
  }

  const unsigned waves_total = (gridDim.x * blockDim.x) >> 5;
  unsigned pair = (blockIdx.x * blockDim.x + threadIdx.x) >> 5;

  for (; pair < total_pairs; pair += waves_total) {
    // Lane l holds x[16t + e] of block (pair*2 + hi): butterfly distances
    // 64/32/16 floats = register strides 4/2/1 -> all intra-lane.
    const float* src = x + (size_t)pair * 256u + (size_t)hi * 128u + e;
    float r[8];
#pragma unroll
    for (int t = 0; t < 8; ++t)
      r[t] = __builtin_nontemporal_load(&src[t * 16]);

    // ---- H8 over chunk index t (unnormalized; scale lives in B) ----
#pragma unroll
    for (int t = 0; t < 4; ++t) {           // stride 4
      float a = r[t], b = r[t + 4];
      r[t] = a + b; r[t + 4] = a - b;
    }
#pragma unroll
    for (int q = 0; q < 8; q += 4)          // stride 2
#pragma unroll
      for (int t = 0; t < 2; ++t) {
        float a = r[q + t], b = r[q + t + 2];
        r[q + t] = a + b; r[q + t + 2] = a - b;
      }
#pragma unroll
    for (int i = 0; i < 8; i += 2) {        // stride 1
      float a = r[i], b = r[i + 1];
      r[i] = a + b; r[i + 1] = a - b;
    }

    // ---- 16x16 transpose via LDS: rows 0-7 = block0 chunks, 8-15 = block1.
    // Store banks: consecutive e -> conflict-free.
#pragma unroll
    for (int t = 0; t < 8; ++t)
      wl[(t + 8 * hi) * PAD + e] = r[t];

    __builtin_amdgcn_wave_barrier();
    asm volatile("s_wait_dscnt 0" ::: "memory");  // wave-private LDS RAW fence
    __builtin_amdgcn_wave_barrier();

    // ---- A fragments + 4 chained f32 WMMAs (K=16) ----
    // f32 WMMA A 16x4 layout: lanes 0-15 row m=lane, K=4j+{0,1};
    // lanes 16-31 row m=lane-16, K=4j+{2,3}.
    v8f acc = {0.f, 0.f, 0.f, 0.f, 0.f, 0.f, 0.f, 0.f};
#pragma unroll
    for (int j = 0; j < 4; ++j) {
      const unsigned base = e * PAD + 4u * (unsigned)j + 2u * hi;
      v2f a;
      a.x = wl[base];       // stride-17 addresses -> conflict-free
      a.y = wl[base + 1u];
      acc = __builtin_amdgcn_wmma_f32_16x16x4_f32(
          /*neg_a=*/false, a, /*neg_b=*/false, bfrag[j],
          /*c_mod=*/(short)0, acc, /*reuse_a=*/false, /*reuse_b=*/false);
    }

    // ---- Store: C/D layout row M=r (lanes 0-15, block0) / M=r+8 (lanes
    // 16-31, block1), N=e. y[16r + e] per block -> same coalesced pattern.
    float* dst = y + (size_t)pair * 256u + (size_t)hi * 128u + e;
#pragma unroll
    for (int t = 0; t < 8; ++t)
      __builtin_nontemporal_store(acc[t], &dst[t * 16]);
  }
}

extern "C" void kernel_launch(void* const* d_in, const int* in_sizes, int n_in,
                              void* d_out, int out_size, void* d_ws, size_t ws_size,
                              hipStream_t stream) {
  const float* x = (const float*)d_in[0];
  const float* H = (const float*)d_in[1];
  float* y = (float*)d_out;
  // one "pair" = 2 blocks of 128 = 256 floats per wave-iteration
  const unsigned total_pairs = (unsigned)(in_sizes[0] / 256);
  dim3 grid(4096), block(256);  // 32768 waves, 8 iterations each
  hipLaunchKernelGGL(fht128_wmma_kernel, grid, block, 0, stream,
                     x, H, y, total_pairs);
}